// DHGAT_3633542332617
// MI455X (gfx1250) — compile-verified
//
#include <hip/hip_runtime.h>
#include <math.h>

#define NNODES 50000
#define NEDGES 400000
#define NH 8
#define ND 32
#define HDIM 256   // H*D

typedef __attribute__((ext_vector_type(2))) float v2f;
typedef __attribute__((ext_vector_type(8))) float v8f;

// ---------------------------------------------------------------------------
// GEMM: Hout[rows,256] = X[rows,K] @ W[K,256], fp32 WMMA 16x16x4.
// One wave (32 threads) per 16(M) x 64(N) output tile; A fragment reused
// across 4 WMMAs per K-step. rows == 50000 (divisible by 16), K % 4 == 0.
// A 16x4 f32 layout: lanes 0-15 hold M=lane, VGPR0/1 = K=0/1; lanes 16-31
// hold the same rows at K=2/3.  B 4x16 mirrors that on the K axis.
// ---------------------------------------------------------------------------
__global__ __launch_bounds__(32) void gemm_wmma_f32(const float* __restrict__ X,
                                                    const float* __restrict__ W,
                                                    float* __restrict__ Hout,
                                                    int K) {
  const int lane = threadIdx.x & 31;
  const int hl   = lane >> 4;    // 0: K+{0,1}, 1: K+{2,3}
  const int l15  = lane & 15;
  const int M0 = blockIdx.x * 16;
  const int N0 = blockIdx.y * 64;
  const long arow = M0 + l15;

  v8f acc0 = {}, acc1 = {}, acc2 = {}, acc3 = {};
  for (int k0 = 0; k0 < K; k0 += 4) {
    const int ka = k0 + 2 * hl;
    v2f a;
    a.x = X[arow * K + ka];
    a.y = X[arow * K + ka + 1];
    const float* wr0 = W + (long)ka * HDIM + N0 + l15;
    const float* wr1 = wr0 + HDIM;
    v2f b0; b0.x = wr0[0];  b0.y = wr1[0];
    v2f b1; b1.x = wr0[16]; b1.y = wr1[16];
    v2f b2; b2.x = wr0[32]; b2.y = wr1[32];
    v2f b3; b3.x = wr0[48]; b3.y = wr1[48];
    acc0 = __builtin_amdgcn_wmma_f32_16x16x4_f32(false, a, false, b0, (short)0, acc0, false, false);
    acc1 = __builtin_amdgcn_wmma_f32_16x16x4_f32(false, a, false, b1, (short)0, acc1, false, false);
    acc2 = __builtin_amdgcn_wmma_f32_16x16x4_f32(false, a, false, b2, (short)0, acc2, false, false);
    acc3 = __builtin_amdgcn_wmma_f32_16x16x4_f32(false, a, false, b3, (short)0, acc3, false, false);
  }
  // C/D layout: VGPR r -> M = r (lanes 0-15) / r+8 (lanes 16-31), N = lane%16
  for (int r = 0; r < 8; ++r) {
    const long row = M0 + r + 8 * hl;
    float* o = Hout + row * HDIM + N0 + l15;
    o[0]  = acc0[r];
    o[16] = acc1[r];
    o[32] = acc2[r];
    o[48] = acc3[r];
  }
}

// Per-(node,head) attention coefficients: as = <h, a_src>, ad = <h, a_dst>
__global__ void attn_coef_kernel(const float* __restrict__ H,
                                 const float* __restrict__ a_src,
                                 const float* __restrict__ a_dst,
                                 float* __restrict__ as_, float* __restrict__ ad_) {
  int t = blockIdx.x * blockDim.x + threadIdx.x;
  if (t >= NNODES * NH) return;
  int n = t >> 3, h = t & 7;
  const float* hp = H + (long)n * HDIM + h * ND;
  const float* sp = a_src + h * ND;
  const float* dp = a_dst + h * ND;
  float s = 0.f, d = 0.f;
  #pragma unroll
  for (int i = 0; i < ND; ++i) { float v = hp[i]; s += v * sp[i]; d += v * dp[i]; }
  as_[t] = s;
  ad_[t] = d;
}

__global__ void fill_kernel(float* __restrict__ p, float v, long n) {
  long i = (long)blockIdx.x * blockDim.x + threadIdx.x;
  if (i < n) p[i] = v;
}

// Per-(edge,head): e = leaky_relu(as[src]+ad[dst], 0.2); segment max via
// int-punned atomic max (m initialized to -inf; self-loops make every
// segment non-empty so the reference's isfinite fixup is a no-op).
__global__ void edge_score_max(const int* __restrict__ src, const int* __restrict__ dst,
                               const float* __restrict__ as_, const float* __restrict__ ad_,
                               float* __restrict__ ebuf, float* __restrict__ m) {
  long t = (long)blockIdx.x * blockDim.x + threadIdx.x;
  const long tot = (long)(NEDGES + NNODES) * NH;
  if (t >= tot) return;
  long e = t >> 3; int h = (int)(t & 7);
  int s = (e < NEDGES) ? src[e] : (int)(e - NEDGES);
  int d = (e < NEDGES) ? dst[e] : (int)(e - NEDGES);
  float v = as_[s * NH + h] + ad_[d * NH + h];
  v = v > 0.f ? v : 0.2f * v;
  ebuf[t] = v;
  float* addr = m + (long)d * NH + h;
  if (v >= 0.f) atomicMax((int*)addr, __float_as_int(v));
  else          atomicMin((unsigned int*)addr, __float_as_uint(v));
}

// Per-(edge,head): ex = exp(e - m[dst]); den[dst] += ex
__global__ void edge_exp_sum(const int* __restrict__ dst, const float* __restrict__ m,
                             float* __restrict__ ebuf, float* __restrict__ den) {
  long t = (long)blockIdx.x * blockDim.x + threadIdx.x;
  const long tot = (long)(NEDGES + NNODES) * NH;
  if (t >= tot) return;
  long e = t >> 3; int h = (int)(t & 7);
  int d = (e < NEDGES) ? dst[e] : (int)(e - NEDGES);
  float ex = __expf(ebuf[t] - m[(long)d * NH + h]);
  ebuf[t] = ex;
  atomicAdd(&den[(long)d * NH + h], ex);
}

// Wave-per-edge weighted scatter: out[dst] += alpha * h[src] (coalesced;
// lane = dim-within-head, heads unrolled). Atomics land on contiguous 128B
// lines that stay resident in the 192MB L2.
__global__ __launch_bounds__(256) void edge_aggregate(const int* __restrict__ src,
                                                      const int* __restrict__ dst,
                                                      const float* __restrict__ ebuf,
                                                      const float* __restrict__ den,
                                                      const float* __restrict__ Hbuf,
                                                      float* __restrict__ Out) {
  int wave = threadIdx.x >> 5;
  int lane = threadIdx.x & 31;
  long e = (long)blockIdx.x * 8 + wave;
  if (e >= NEDGES + NNODES) return;
  int s = (e < NEDGES) ? src[e] : (int)(e - NEDGES);
  int d = (e < NEDGES) ? dst[e] : (int)(e - NEDGES);
  const float* hs = Hbuf + (long)s * HDIM;
  float*       op = Out  + (long)d * HDIM;
  #pragma unroll
  for (int j = 0; j < NH; ++j) {
    float alpha = ebuf[e * NH + j] / (den[(long)d * NH + j] + 1e-16f);
    atomicAdd(op + j * ND + lane, alpha * hs[j * ND + lane]);
  }
}

// out = elu(in + bias)
__global__ void bias_elu_kernel(const float* __restrict__ In, const float* __restrict__ b,
                                float* __restrict__ Out, long n) {
  long i = (long)blockIdx.x * blockDim.x + threadIdx.x;
  if (i >= n) return;
  float v = In[i] + b[i & (HDIM - 1)];
  Out[i] = v > 0.f ? v : (__expf(v) - 1.f);
}

// Dual-channel attention + MLP head + log_softmax, one wave per node.
__global__ __launch_bounds__(256) void head_kernel(const float* __restrict__ C,
                                                   const float* __restrict__ S,
                                                   const float* __restrict__ attc_w,
                                                   const float* __restrict__ attc_b,
                                                   const float* __restrict__ atts_w,
                                                   const float* __restrict__ atts_b,
                                                   const float* __restrict__ fc1_w,
                                                   const float* __restrict__ fc1_b,
                                                   const float* __restrict__ fc2_w,
                                                   const float* __restrict__ fc2_b,
                                                   float* __restrict__ out) {
  int wave = threadIdx.x >> 5;
  int lane = threadIdx.x & 31;
  int n = blockIdx.x * 8 + wave;
  if (n >= NNODES) return;
  const float* c = C + (long)n * HDIM;
  const float* s = S + (long)n * HDIM;

  float cd = 0.f, sd = 0.f;
  #pragma unroll
  for (int j = 0; j < 8; ++j) {
    int i = lane + 32 * j;
    cd += c[i] * attc_w[i];
    sd += s[i] * atts_w[i];
  }
  #pragma unroll
  for (int off = 16; off; off >>= 1) {
    cd += __shfl_xor(cd, off, 32);
    sd += __shfl_xor(sd, off, 32);
  }
  float cs = 1.f / (1.f + __expf(-(cd + attc_b[0])));
  float ss = 1.f / (1.f + __expf(-(sd + atts_b[0])));
  float mx = fmaxf(cs, ss);
  float e0 = __expf(cs - mx), e1 = __expf(ss - mx);
  float inv = 1.f / (e0 + e1);
  float w0 = e0 * inv, w1 = e1 * inv;

  // fc1: lane computes output feature `lane` (fc1_w is [512,32] row-major)
  float y = fc1_b[lane];
  for (int i = 0; i < HDIM; ++i) y += (w0 * c[i]) * fc1_w[i * 32 + lane];
  for (int i = 0; i < HDIM; ++i) y += (w1 * s[i]) * fc1_w[(HDIM + i) * 32 + lane];
  y = y > 0.f ? y : (__expf(y) - 1.f);

  float p0 = y * fc2_w[lane * 2 + 0];
  float p1 = y * fc2_w[lane * 2 + 1];
  #pragma unroll
  for (int off = 16; off; off >>= 1) {
    p0 += __shfl_xor(p0, off, 32);
    p1 += __shfl_xor(p1, off, 32);
  }
  if (lane == 0) {
    float z0 = p0 + fc2_b[0], z1 = p1 + fc2_b[1];
    float zm = fmaxf(z0, z1);
    float lse = zm + __logf(__expf(z0 - zm) + __expf(z1 - zm));
    out[n * 2 + 0] = z0 - lse;
    out[n * 2 + 1] = z1 - lse;
  }
}

// ---------------------------------------------------------------------------

extern "C" void kernel_launch(void* const* d_in, const int* in_sizes, int n_in,
                              void* d_out, int out_size, void* d_ws, size_t ws_size,
                              hipStream_t stream) {
  const float* content_x = (const float*)d_in[0];
  const float* social_x  = (const float*)d_in[1];
  const int* cei = (const int*)d_in[2];
  const int* sei = (const int*)d_in[3];
  const int *csrc = cei, *cdst = cei + NEDGES;
  const int *ssrc = sei, *sdst = sei + NEDGES;
  const float* Wc0  = (const float*)d_in[4];
  const float* ac0s = (const float*)d_in[5];
  const float* ac0d = (const float*)d_in[6];
  const float* bc0  = (const float*)d_in[7];
  const float* Wc1  = (const float*)d_in[8];
  const float* ac1s = (const float*)d_in[9];
  const float* ac1d = (const float*)d_in[10];
  const float* bc1  = (const float*)d_in[11];
  const float* Ws0  = (const float*)d_in[12];
  const float* as0s = (const float*)d_in[13];
  const float* as0d = (const float*)d_in[14];
  const float* bs0  = (const float*)d_in[15];
  const float* Ws1  = (const float*)d_in[16];
  const float* as1s = (const float*)d_in[17];
  const float* as1d = (const float*)d_in[18];
  const float* bs1  = (const float*)d_in[19];
  const float* attc_w = (const float*)d_in[20];
  const float* attc_b = (const float*)d_in[21];
  const float* atts_w = (const float*)d_in[22];
  const float* atts_b = (const float*)d_in[23];
  const float* fc1_w  = (const float*)d_in[24];
  const float* fc1_b  = (const float*)d_in[25];
  const float* fc2_w  = (const float*)d_in[26];
  const float* fc2_b  = (const float*)d_in[27];

  // Workspace carve-up (floats)
  float* ws = (float*)d_ws;
  float* hbuf  = ws; ws += (long)NNODES * HDIM;   // transformed features h
  float* bufA  = ws; ws += (long)NNODES * HDIM;   // layer ping-pong
  float* bufB  = ws; ws += (long)NNODES * HDIM;   // final content channel
  float* bufC  = ws; ws += (long)NNODES * HDIM;   // final social channel
  float* asbuf = ws; ws += (long)NNODES * NH;
  float* adbuf = ws; ws += (long)NNODES * NH;
  float* mbuf  = ws; ws += (long)NNODES * NH;
  float* denbuf= ws; ws += (long)NNODES * NH;
  float* ebuf  = ws; ws += (long)(NEDGES + NNODES) * NH;

  const long nhTot   = (long)NNODES * NH;
  const long featTot = (long)NNODES * HDIM;
  const long edgeTot = (long)(NEDGES + NNODES) * NH;

  auto run_layer = [&](const float* X, int K, const float* W,
                       const float* aw_s, const float* aw_d, const float* bias,
                       const int* src, const int* dst, float* outbuf) {
    gemm_wmma_f32<<<dim3(NNODES / 16, HDIM / 64), 32, 0, stream>>>(X, W, hbuf, K);
    attn_coef_kernel<<<(int)((nhTot + 255) / 256), 256, 0, stream>>>(hbuf, aw_s, aw_d, asbuf, adbuf);
    fill_kernel<<<(int)((nhTot + 255) / 256), 256, 0, stream>>>(mbuf, -INFINITY, nhTot);
    fill_kernel<<<(int)((nhTot + 255) / 256), 256, 0, stream>>>(denbuf, 0.f, nhTot);
    fill_kernel<<<(int)((featTot + 255) / 256), 256, 0, stream>>>(outbuf, 0.f, featTot);
    edge_score_max<<<(int)((edgeTot + 255) / 256), 256, 0, stream>>>(src, dst, asbuf, adbuf, ebuf, mbuf);
    edge_exp_sum<<<(int)((edgeTot + 255) / 256), 256, 0, stream>>>(dst, mbuf, ebuf, denbuf);
    edge_aggregate<<<(NEDGES + NNODES + 7) / 8, 256, 0, stream>>>(src, dst, ebuf, denbuf, hbuf, outbuf);
    bias_elu_kernel<<<(int)((featTot + 255) / 256), 256, 0, stream>>>(outbuf, bias, outbuf, featTot);
  };

  // content channel: 768 -> 256 -> 256
  run_layer(content_x, 768, Wc0, ac0s, ac0d, bc0, csrc, cdst, bufA);
  run_layer(bufA,      256, Wc1, ac1s, ac1d, bc1, csrc, cdst, bufB);
  // social channel: 128 -> 256 -> 256
  run_layer(social_x,  128, Ws0, as0s, as0d, bs0, ssrc, sdst, bufA);
  run_layer(bufA,      256, Ws1, as1s, as1d, bs1, ssrc, sdst, bufC);

  head_kernel<<<(NNODES + 7) / 8, 256, 0, stream>>>(bufB, bufC, attc_w, attc_b,
                                                    atts_w, atts_b, fc1_w, fc1_b,
                                                    fc2_w, fc2_b, (float*)d_out);
}